// OccupancyGridForestAS_53077205844615
// MI455X (gfx1250) — compile-verified
//
#include <hip/hip_runtime.h>

#define N_PTS   4194304
#define RES     64
#define LDIM    8
#define THREADS 256
#define PTS_PER_THREAD 4

typedef float v4f __attribute__((ext_vector_type(4)));

__global__ __launch_bounds__(THREADS) void OccupancyGridForestAS_kernel(
    const float* __restrict__ pts,        // [N_PTS, 3]
    const float* __restrict__ grid,       // [64, 64, 64, 64]
    const int*   __restrict__ lookup,     // [8, 8, 8]
    float*       __restrict__ out)        // [N_PTS]
{
    __shared__ int s_lookup[LDIM * LDIM * LDIM]; // 512 entries = 2 KB

    const int t   = threadIdx.x;
    const int gid = blockIdx.x * THREADS + t;       // one thread = 4 points
    const int base = gid * PTS_PER_THREAD;

    // Prefetch this thread's 48-byte pts chunk (spans up to 2 cachelines)
    // BEFORE the LDS staging + barrier, so the global loads after the
    // barrier hit cache. Emits global_prefetch_b8.
    if (base < N_PTS) {
        const char* pbase = (const char*)(pts + (size_t)base * 3);
        __builtin_prefetch(pbase, 0, 3);
        __builtin_prefetch(pbase + 47, 0, 3);
    }

    // Stage the 2 KB lookup table into LDS: 512 entries, 256 threads.
    s_lookup[t]           = lookup[t];
    s_lookup[t + THREADS] = lookup[t + THREADS];
    __syncthreads();   // lowers to s_barrier_signal / s_barrier_wait

    if (base >= N_PTS) return;

    // 4 points = 12 floats = 3 x float4, 16B-aligned (base*12B, base%4==0).
    // Non-temporal: pts is streamed once; keep L2 free for the 64MB grid.
    const v4f* p4 = reinterpret_cast<const v4f*>(pts + (size_t)base * 3);
    v4f a = __builtin_nontemporal_load(p4 + 0);
    v4f b = __builtin_nontemporal_load(p4 + 1);
    v4f c = __builtin_nontemporal_load(p4 + 2);

    const float px[4] = {a.x, a.w, b.z, c.y};
    const float py[4] = {a.y, b.x, b.w, c.z};
    const float pz[4] = {a.z, b.y, c.x, c.w};

    v4f res;

#pragma unroll
    for (int k = 0; k < PTS_PER_THREAD; ++k) {
        const float x = px[k], y = py[k], z = pz[k];

        const float fx = floorf(x), fy = floorf(y), fz = floorf(z);
        const int bx = (int)fx, by = (int)fy, bz = (int)fz;

        const bool in_dom = (bx >= 0) & (bx < LDIM) &
                            (by >= 0) & (by < LDIM) &
                            (bz >= 0) & (bz < LDIM);

        const int bxs = min(max(bx, 0), LDIM - 1);
        const int bys = min(max(by, 0), LDIM - 1);
        const int bzs = min(max(bz, 0), LDIM - 1);

        const int bidx  = s_lookup[(bxs * LDIM + bys) * LDIM + bzs]; // ds_load_b32
        const bool valid = in_dom && (bidx >= 0);

        // frac = pts - clamped_cell  (== (block_x*0.5+0.5) in the reference)
        const float frx = x - (float)bxs;
        const float fry = y - (float)bys;
        const float frz = z - (float)bzs;

        const int vx = min(max((int)floorf(frx * (float)RES), 0), RES - 1);
        const int vy = min(max((int)floorf(fry * (float)RES), 0), RES - 1);
        const int vz = min(max((int)floorf(frz * (float)RES), 0), RES - 1);

        const int sb = valid ? bidx : 0;
        const size_t idx = (((size_t)sb * RES + vx) * RES + vy) * RES + vz;

        // RT-policy gather: 64MB grid stays resident in the 192MB L2.
        const float v = grid[idx];
        res[k] = valid ? v : 0.0f;
    }

    // Non-temporal streaming store: out is never re-read by this kernel.
    __builtin_nontemporal_store(res, reinterpret_cast<v4f*>(out) + gid);
}

extern "C" void kernel_launch(void* const* d_in, const int* in_sizes, int n_in,
                              void* d_out, int out_size, void* d_ws, size_t ws_size,
                              hipStream_t stream) {
    const float* pts    = (const float*)d_in[0];  // 4194304*3 f32
    const float* grid   = (const float*)d_in[1];  // 64*64*64*64 f32
    const int*   lookup = (const int*)d_in[2];    // 8*8*8 i32
    float*       out    = (float*)d_out;          // 4194304 f32

    const int n_threads = N_PTS / PTS_PER_THREAD;       // 1,048,576
    const int n_blocks  = n_threads / THREADS;          // 4096

    OccupancyGridForestAS_kernel<<<n_blocks, THREADS, 0, stream>>>(pts, grid, lookup, out);
}